// GraphSAGEEncoder_85736137163071
// MI455X (gfx1250) — compile-verified
//
#include <hip/hip_runtime.h>
#include <hip/hip_bf16.h>

typedef float v2f __attribute__((ext_vector_type(2)));
typedef float v8f __attribute__((ext_vector_type(8)));

#define N_NODES 50000
#define N_EDGES 800000
// IN_C == HID_C == 128, OUT_C == 64

// Padded LDS row stride (floats). 132 => row offset mod 64 banks = 4*row, so
// the 16 "M" lanes of a WMMA A-fragment b64 load hit 16 distinct bank pairs,
// and the upper lane half (K+2) hits a disjoint set -> conflict-free.
#define LDS_STRIDE 132
#define MAX_ROWS_PER_BLOCK 32

// ---------------------------------------------------------------------------
// Relaxed device-scope float atomic add -> GLOBAL_ATOMIC_ADD_F32 (no CAS loop)
// ---------------------------------------------------------------------------
__device__ __forceinline__ void atomic_add_f32(float* p, float v) {
    __hip_atomic_fetch_add(p, v, __ATOMIC_RELAXED, __HIP_MEMORY_SCOPE_AGENT);
}

// ---------------------------------------------------------------------------
// Edge scatter: one wave32 per edge; lane c handles channels {c, c+32, c+64,
// c+96}. Gather of feat[src] is fully coalesced (128 consecutive floats per
// wave); scatter is 4 f32 atomics per lane into the L2-resident accumulator.
// ---------------------------------------------------------------------------
__global__ __launch_bounds__(256) void scatter_sum_kernel(
    const float* __restrict__ feat,   // [N, 128]
    const int*   __restrict__ src,    // [E]
    const int*   __restrict__ dst,    // [E]
    float*       __restrict__ s,      // [N, 128] accumulator (pre-zeroed)
    float*       __restrict__ cnt,    // [N] degree accumulator (pre-zeroed)
    int n_edges, int accum_cnt)
{
    const int lane = threadIdx.x & 31;
    const int e    = blockIdx.x * 8 + (threadIdx.x >> 5);   // wave-uniform
    if (e >= n_edges) return;

    const int sn = src[e];
    const int dn = dst[e];
    const float* xs = feat + (size_t)sn * 128;
    float*       sd = s    + (size_t)dn * 128;

#pragma unroll
    for (int i = 0; i < 4; ++i) {
        const int c = lane + i * 32;
        atomic_add_f32(&sd[c], xs[c]);
    }
    if (accum_cnt && lane == 0) atomic_add_f32(&cnt[dn], 1.0f);
}

// ---------------------------------------------------------------------------
// inv[n] = 1 / max(cnt[n], 1)
// ---------------------------------------------------------------------------
__global__ __launch_bounds__(256) void inv_cnt_kernel(
    const float* __restrict__ cnt, float* __restrict__ inv, int n)
{
    const int i = blockIdx.x * blockDim.x + threadIdx.x;
    if (i < n) inv[i] = 1.0f / fmaxf(cnt[i], 1.0f);
}

// ---------------------------------------------------------------------------
// Fused SAGE dense stage:
//   out[n, :] = act( (s[n,:] * inv[n]) @ Wl  +  x[n,:] @ Wr  +  bias )
//
// Block = 256 threads (8 waves). The block's A rows (s and x) are staged into
// LDS ONCE via CDNA5 async global->LDS copies (b128, ASYNCcnt), because all
// col-tile waves of the block consume the SAME rows (8x / 4x reuse). WMMA A
// fragments are then served from LDS (ds_load_b64, conflict-free via padded
// stride); B fragments stream from the L2-resident weight matrices.
//
// One wave32 per 16x16 output tile, V_WMMA_F32_16X16X4_F32 over K=128.
// Fragment layouts per CDNA5 ISA 7.12.2 (f32, wave32). All WMMA-region
// branches are wave-uniform => EXEC all-1s at every WMMA (ISA requirement).
// ---------------------------------------------------------------------------
__global__ __launch_bounds__(256) void sage_wmma_kernel(
    const float* __restrict__ s,     // [N, 128] neighbor sums
    const float* __restrict__ inv,   // [N]
    const float* __restrict__ x,     // [N, 128] self features
    const float* __restrict__ Wl,    // [128, ncols]
    const float* __restrict__ Wr,    // [128, ncols]
    const float* __restrict__ bias,  // [ncols]
    float*       __restrict__ out,   // [N, ncols]
    int n_rows, int ncols, int do_relu)
{
    __shared__ float lds_s[MAX_ROWS_PER_BLOCK * LDS_STRIDE];
    __shared__ float lds_x[MAX_ROWS_PER_BLOCK * LDS_STRIDE];

    const int lane  = threadIdx.x & 31;
    const int wave  = threadIdx.x >> 5;
    const int colTiles         = ncols >> 4;        // 8 (hid) or 4 (out)
    const int rowTilesPerBlock = 8 / colTiles;      // 1 or 2
    const int rowsPerBlock     = rowTilesPerBlock * 16;
    const int rowBase          = blockIdx.x * rowsPerBlock;

    // ---- Stage A rows (s and x) into LDS with async b128 copies ----
    // chunk = 16B of one row; 32 chunks per 128-float row.
    const int totalChunks = rowsPerBlock * 32;      // per matrix
    for (int c = threadIdx.x; c < totalChunks; c += 256) {
        const int r    = c >> 5;                    // row within block
        const int ch   = c & 31;                    // float4 chunk within row
        const int grow = rowBase + r;
        if (grow < n_rows) {
            const unsigned lds_off_s =
                (unsigned)(unsigned long long)(void*)&lds_s[r * LDS_STRIDE + ch * 4];
            const unsigned lds_off_x =
                (unsigned)(unsigned long long)(void*)&lds_x[r * LDS_STRIDE + ch * 4];
            const unsigned long long gs =
                (unsigned long long)(const void*)(s + (size_t)grow * 128 + ch * 4);
            const unsigned long long gx =
                (unsigned long long)(const void*)(x + (size_t)grow * 128 + ch * 4);
            asm volatile("global_load_async_to_lds_b128 %0, %1, off"
                         :: "v"(lds_off_s), "v"(gs) : "memory");
            asm volatile("global_load_async_to_lds_b128 %0, %1, off"
                         :: "v"(lds_off_x), "v"(gx) : "memory");
        }
    }
    asm volatile("s_wait_asynccnt 0" ::: "memory");
    __syncthreads();

    // ---- Per-wave 16x16 WMMA tile ----
    const int colTile = wave % colTiles;
    const int rowTile = blockIdx.x * rowTilesPerBlock + wave / colTiles;

    if (rowTile * 16 < n_rows) {                    // wave-uniform guard
        const int m      = lane & 15;               // A row / B,D column
        const int khalf  = (lane >> 4) << 1;        // 0 or 2
        const int rLocal = (wave / colTiles) * 16 + m;
        const int col0   = colTile * 16;

        const float  inv_r = inv[rowTile * 16 + m];
        const float* lsr   = &lds_s[rLocal * LDS_STRIDE];
        const float* lxr   = &lds_x[rLocal * LDS_STRIDE];

        v8f acc = {0.f, 0.f, 0.f, 0.f, 0.f, 0.f, 0.f, 0.f};

#pragma unroll 4
        for (int k0 = 0; k0 < 128; k0 += 4) {
            const int kk = k0 + khalf;

            v2f a_mean = { lsr[kk] * inv_r, lsr[kk + 1] * inv_r };
            v2f a_self = { lxr[kk],         lxr[kk + 1] };
            v2f b_l = { Wl[(size_t)kk * ncols + col0 + m],
                        Wl[(size_t)(kk + 1) * ncols + col0 + m] };
            v2f b_r = { Wr[(size_t)kk * ncols + col0 + m],
                        Wr[(size_t)(kk + 1) * ncols + col0 + m] };

            acc = __builtin_amdgcn_wmma_f32_16x16x4_f32(
                      false, a_mean, false, b_l, (short)0, acc, false, false);
            acc = __builtin_amdgcn_wmma_f32_16x16x4_f32(
                      false, a_self, false, b_r, (short)0, acc, false, false);
        }

        const int mbase  = rowTile * 16 + ((lane >> 4) << 3);  // +0 or +8
        const int outcol = col0 + m;
        const float b = bias[outcol];
#pragma unroll
        for (int r = 0; r < 8; ++r) {
            float v = acc[r] + b;
            if (do_relu) v = fmaxf(v, 0.f);
            out[(size_t)(mbase + r) * ncols + outcol] = v;
        }
    }
}

// ---------------------------------------------------------------------------
// Host-side launch (graph-capture safe: only async ops on `stream`)
// ---------------------------------------------------------------------------
extern "C" void kernel_launch(void* const* d_in, const int* in_sizes, int n_in,
                              void* d_out, int out_size, void* d_ws, size_t ws_size,
                              hipStream_t stream)
{
    (void)in_sizes; (void)n_in; (void)out_size; (void)ws_size;

    const float* x    = (const float*)d_in[0];
    const int*   ei   = (const int*)  d_in[1];
    const float* W_l0 = (const float*)d_in[2];
    const float* b0   = (const float*)d_in[3];
    const float* W_r0 = (const float*)d_in[4];
    const float* W_l1 = (const float*)d_in[5];
    const float* b1   = (const float*)d_in[6];
    const float* W_r1 = (const float*)d_in[7];

    const int* src = ei;             // edge_index row 0
    const int* dst = ei + N_EDGES;   // edge_index row 1

    // Workspace carve-out (256B aligned slabs)
    char*  ws  = (char*)d_ws;
    size_t off = 0;
    auto carve = [&](size_t nfloats) {
        float* p = (float*)(ws + off);
        off += ((nfloats * sizeof(float)) + 255) & ~(size_t)255;
        return p;
    };
    float* s   = carve((size_t)N_NODES * 128);  // reused by both layers
    float* cnt = carve(N_NODES);
    float* inv = carve(N_NODES);
    float* h   = carve((size_t)N_NODES * 128);

    const int edgeBlocks = (N_EDGES + 7) / 8;        // 8 edges (waves) / block
    const int rowTiles   = N_NODES / 16;             // 50000/16 = 3125 exact

    // ----- Layer 0 -----
    hipMemsetAsync(s,   0, (size_t)N_NODES * 128 * sizeof(float), stream);
    hipMemsetAsync(cnt, 0, (size_t)N_NODES * sizeof(float),       stream);
    scatter_sum_kernel<<<edgeBlocks, 256, 0, stream>>>(x, src, dst, s, cnt,
                                                       N_EDGES, 1);
    inv_cnt_kernel<<<(N_NODES + 255) / 256, 256, 0, stream>>>(cnt, inv, N_NODES);
    // h = relu(mean @ W_l0 + x @ W_r0 + b0); 128 cols -> 8 col-tiles, 16 rows/block
    sage_wmma_kernel<<<rowTiles, 256, 0, stream>>>(s, inv, x, W_l0, W_r0, b0,
                                                   h, N_NODES, 128, 1);

    // ----- Layer 1 -----
    hipMemsetAsync(s, 0, (size_t)N_NODES * 128 * sizeof(float), stream);
    scatter_sum_kernel<<<edgeBlocks, 256, 0, stream>>>(h, src, dst, s, cnt,
                                                       N_EDGES, 0);
    // out = mean_h @ W_l1 + h @ W_r1 + b1; 64 cols -> 4 col-tiles, 32 rows/block
    sage_wmma_kernel<<<(rowTiles + 1) / 2, 256, 0, stream>>>(
        s, inv, h, W_l1, W_r1, b1, (float*)d_out, N_NODES, 64, 0);
}